// PositionalEncoding_35553739276819
// MI455X (gfx1250) — compile-verified
//
#include <hip/hip_runtime.h>
#include <hip/hip_bf16.h>
#include <math.h>

// ---------------------------------------------------------------------------
// PositionalEncoding fused kernels for MI455X (gfx1250, wave32, WMMA)
//
// Rewrite: segment_mean(pe[row]) with pe = nan_to_num(graph_lpe)@pe_w + pe_b
// == (segment_mean(nan_to_num(graph_lpe)[row])) @ pe_w + (cnt>0)*pe_b
// -> aggregate 16-dim raw LPE per clique (half the scatter bytes; hot tables
//    graph_lpe 128MB + agg16 64MB + cnt 4MB ~ fit 192MB L2), project once.
// Dense math via V_WMMA_F32_16X16X4_F32 (fp32-exact vs ref).
//
// Round-2 tuning: pair-interleaved B weights in LDS (1x ds_load_b64 per
// fragment, no v_mov glue), bank-conflict-free padded strides (A tile: 68
// floats -> 4m bank rotation; 64-wide B pairs: 80 float2 -> +32 bank offset
// for the high half-wave), and A-fragments hoisted to registers (loaded once
// per GEMM instead of once per n-tile).
// ---------------------------------------------------------------------------

typedef __attribute__((ext_vector_type(2))) float v2f;
typedef __attribute__((ext_vector_type(8))) float v8f;

#define WAVES_PER_BLOCK 4
#define TILE_M 16
#define ALD 68   // A-tile row stride (floats): conflict-free, 16B-aligned rows
#define TLD 20   // 16-wide A-tile row stride (floats)
#define W64S 80  // pair-row stride (float2) for 64x64 weights: half-wave +32 banks
#define W16S 32  // pair-row stride (float2) for 16x32 weights

__device__ __forceinline__ float nan_fix(float v) {
  if (__builtin_isnan(v)) return 0.0f;
  if (__builtin_isinf(v)) return v > 0.0f ? 3.402823466e38f : -3.402823466e38f;
  return v;
}

__device__ __forceinline__ v8f wmma4(v2f a, v2f b, v8f c) {
  // D = A(16x4 f32) * B(4x16 f32) + C(16x16 f32)
  return __builtin_amdgcn_wmma_f32_16x16x4_f32(
      /*neg_a=*/false, a, /*neg_b=*/false, b,
      /*c_mod=*/(short)0, c, /*reuse_a=*/false, /*reuse_b=*/false);
}

// A fragment from a row-major 16xK LDS tile (leading dim lda, floats).
// ISA layout: lanes 0-15 hold M=lane, K={kk,kk+1}; lanes 16-31 hold
// M=lane-16, K={kk+2,kk+3}.  kb even -> one aligned ds_load_b64.
__device__ __forceinline__ v2f load_a2(const float* A, int lane, int kk, int lda) {
  const int m = lane & 15;
  const int kb = kk + ((lane >> 4) << 1);
  return *(const v2f*)&A[m * lda + kb];
}

// B fragment from pair-interleaved LDS weights:
// Wp[kp*stride + n] (as float2) = {W[2kp][n], W[2kp+1][n]}.
// lane needs pair kp = kk/2 + half -> single aligned ds_load_b64.
__device__ __forceinline__ v2f load_bp(const float* Wp, int lane, int kk, int n0,
                                       int stride_pairs) {
  const int n = n0 + (lane & 15);
  const int kp = (kk >> 1) + (lane >> 4);
  return *(const v2f*)&Wp[(kp * stride_pairs + n) * 2];
}

// ---------------------------------------------------------------------------
__global__ void zero_ws_kernel(float* __restrict__ p, size_t n) {
  const size_t i = (size_t)blockIdx.x * blockDim.x + threadIdx.x;
  if (i < n) p[i] = 0.0f;
}

// 16 lanes per edge: coalesced 64B gather of graph_lpe[row], scatter-add
// into agg16[col] (+ cnt). L2-resident tables -> atomics stay on-chip.
__global__ void edge_scatter_kernel(const int* __restrict__ row,
                                    const int* __restrict__ col,
                                    const float* __restrict__ graph_lpe,
                                    float* __restrict__ agg16,
                                    float* __restrict__ cnt,
                                    int n_edge) {
  const size_t t = (size_t)blockIdx.x * blockDim.x + threadIdx.x;
  const size_t e = t >> 4;
  const int f = (int)(t & 15);
  if (e >= (size_t)n_edge) return;
  const int r = row[e];
  const int c = col[e];
  const float v = nan_fix(graph_lpe[(size_t)r * 16 + f]);
  atomicAdd(&agg16[(size_t)c * 16 + f], v);
  if (f == 0) atomicAdd(&cnt[c], 1.0f);
}

// ---------------------------------------------------------------------------
// One wave per 16-clique-row tile. All four matmuls via WMMA f32 16x16x4.
__global__ __launch_bounds__(WAVES_PER_BLOCK * 32)
void clique_fused_kernel(const float* __restrict__ x_clique,
                         const int* __restrict__ tree_degree,
                         const float* __restrict__ tree_lpe,
                         const float* __restrict__ deg_emb,
                         const float* __restrict__ deg_w,
                         const float* __restrict__ deg_b,
                         const float* __restrict__ merge_w,
                         const float* __restrict__ merge_b,
                         const float* __restrict__ tree_w,
                         const float* __restrict__ tree_b,
                         const float* __restrict__ pe_w,
                         const float* __restrict__ pe_b,
                         const float* __restrict__ agg16,
                         const float* __restrict__ cnt,
                         float* __restrict__ out,
                         int n_clique) {
  // pair-interleaved weights (see load_bp)
  __shared__ __align__(16) float s_deg_w[32 * W64S * 2];    // 20 KB
  __shared__ __align__(16) float s_merge_w[32 * W64S * 2];  // 20 KB
  __shared__ __align__(16) float s_tree_w[8 * W16S * 2];    // 2 KB
  __shared__ __align__(16) float s_pe_w[8 * W16S * 2];      // 2 KB
  __shared__ __align__(16) float s_atile[WAVES_PER_BLOCK][16 * ALD];  // 4.25 KB/wave

  const int tid = threadIdx.x;
  for (int i = tid; i < 64 * 64; i += (int)blockDim.x) {
    const int k = i >> 6, n = i & 63;
    const int off = ((k >> 1) * W64S + n) * 2 + (k & 1);
    s_deg_w[off] = deg_w[i];
    s_merge_w[off] = merge_w[i];
  }
  for (int i = tid; i < 16 * 32; i += (int)blockDim.x) {
    const int k = i >> 5, n = i & 31;
    const int off = ((k >> 1) * W16S + n) * 2 + (k & 1);
    s_tree_w[off] = tree_w[i];
    s_pe_w[off] = pe_w[i];
  }
  __syncthreads();

  const int wave = tid >> 5;
  const int lane = tid & 31;
  const int row0 = (blockIdx.x * WAVES_PER_BLOCK + wave) * TILE_M;
  if (row0 >= n_clique) return;  // wave-uniform: EXEC stays all-ones for WMMA

  float* A = s_atile[wave];  // private per-wave staging tile
  const int m16 = lane & 15;
  const int half = lane >> 4;
  const int mrow = min(row0 + m16, n_clique - 1);

  // ---- stage degree-embedding rows: A[m][0:64] = deg_emb[tree_degree[m]] ----
  {
    const int d = tree_degree[mrow];
    const float4* src = (const float4*)(deg_emb + (size_t)d * 64 + half * 32);
    float4* dst = (float4*)(A + m16 * ALD + half * 32);  // rows 16B-aligned (ALD=68)
#pragma unroll
    for (int k = 0; k < 8; ++k) dst[k] = src[k];
  }

  // ---- GEMM1: h1 = relu(deg @ deg_w + deg_b) + x_clique  (16x64) ----
  v2f af[16];
#pragma unroll
  for (int k4 = 0; k4 < 16; ++k4) af[k4] = load_a2(A, lane, k4 * 4, ALD);

  v8f h1[4];
#pragma unroll
  for (int nn = 0; nn < 4; ++nn) {
    v8f acc = {};
#pragma unroll
    for (int k4 = 0; k4 < 16; ++k4)
      acc = wmma4(af[k4], load_bp(s_deg_w, lane, k4 * 4, nn * 16, W64S), acc);
    const int n = nn * 16 + m16;
    const float bias = deg_b[n];
#pragma unroll
    for (int v = 0; v < 8; ++v) {
      const int r = min(row0 + v + half * 8, n_clique - 1);
      float t = acc[v] + bias;
      t = t > 0.0f ? t : 0.0f;
      h1[nn][v] = t + x_clique[(size_t)r * 64 + n];
    }
  }

  // ---- restage h1 into LDS tile (row-major 16xALD); same-wave DS in-order ----
#pragma unroll
  for (int nn = 0; nn < 4; ++nn) {
    const int n = nn * 16 + m16;
#pragma unroll
    for (int v = 0; v < 8; ++v) A[(v + half * 8) * ALD + n] = h1[nn][v];
  }

  // ---- GEMM2: xc = h1 @ merge_w + merge_b ----
#pragma unroll
  for (int k4 = 0; k4 < 16; ++k4) af[k4] = load_a2(A, lane, k4 * 4, ALD);

  v8f xc[4];
#pragma unroll
  for (int nn = 0; nn < 4; ++nn) {
    v8f acc = {};
#pragma unroll
    for (int k4 = 0; k4 < 16; ++k4)
      acc = wmma4(af[k4], load_bp(s_merge_w, lane, k4 * 4, nn * 16, W64S), acc);
    const float bias = merge_b[nn * 16 + m16];
#pragma unroll
    for (int v = 0; v < 8; ++v) xc[nn][v] = acc[v] + bias;
  }

  // ---- tree_pe = nan_to_num(tree_lpe) @ tree_w + tree_b  (16x32) ----
  {
    const float* src = tree_lpe + (size_t)mrow * 16 + half * 8;
    float* dst = A + m16 * TLD + half * 8;
#pragma unroll
    for (int k = 0; k < 8; ++k) dst[k] = nan_fix(src[k]);
  }
  v2f af4[4];
#pragma unroll
  for (int k4 = 0; k4 < 4; ++k4) af4[k4] = load_a2(A, lane, k4 * 4, TLD);

  v8f tp[2];
#pragma unroll
  for (int nn = 0; nn < 2; ++nn) {
    v8f acc = {};
#pragma unroll
    for (int k4 = 0; k4 < 4; ++k4)
      acc = wmma4(af4[k4], load_bp(s_tree_w, lane, k4 * 4, nn * 16, W16S), acc);
    const float bias = tree_b[nn * 16 + m16];
#pragma unroll
    for (int v = 0; v < 8; ++v) tp[nn][v] = acc[v] + bias;
  }

  // ---- pe_agg = (agg16/max(cnt,1)) @ pe_w + (cnt>0)*pe_b  (16x32) ----
  {
    const float c = cnt[mrow];
    const float inv = 1.0f / fmaxf(c, 1.0f);
    const float* src = agg16 + (size_t)mrow * 16 + half * 8;
    float* dst = A + m16 * TLD + half * 8;
#pragma unroll
    for (int k = 0; k < 8; ++k) dst[k] = src[k] * inv;
  }
#pragma unroll
  for (int k4 = 0; k4 < 4; ++k4) af4[k4] = load_a2(A, lane, k4 * 4, TLD);

  v8f pa[2];
#pragma unroll
  for (int nn = 0; nn < 2; ++nn) {
    v8f acc = {};
#pragma unroll
    for (int k4 = 0; k4 < 4; ++k4)
      acc = wmma4(af4[k4], load_bp(s_pe_w, lane, k4 * 4, nn * 16, W16S), acc);
    const float bias = pe_b[nn * 16 + m16];
#pragma unroll
    for (int v = 0; v < 8; ++v) {
      const int r = min(row0 + v + half * 8, n_clique - 1);
      const float mask = cnt[r] > 0.0f ? 1.0f : 0.0f;
      pa[nn][v] = acc[v] + bias * mask;
    }
  }

  // ---- epilogue: out = xc + concat([pe_agg, tree_pe], axis=-1) ----
#pragma unroll
  for (int nn = 0; nn < 4; ++nn) {
    const int n = nn * 16 + m16;
#pragma unroll
    for (int v = 0; v < 8; ++v) {
      const int r = row0 + v + half * 8;
      if (r < n_clique) {
        const float addv = (nn < 2) ? pa[nn][v] : tp[nn - 2][v];
        out[(size_t)r * 64 + n] = xc[nn][v] + addv;
      }
    }
  }
}

// ---------------------------------------------------------------------------
extern "C" void kernel_launch(void* const* d_in, const int* in_sizes, int n_in,
                              void* d_out, int out_size, void* d_ws, size_t ws_size,
                              hipStream_t stream) {
  const float* x_clique   = (const float*)d_in[0];
  const int*   tree_deg   = (const int*)d_in[1];
  const float* tree_lpe   = (const float*)d_in[2];
  const float* graph_lpe  = (const float*)d_in[3];
  const int*   row        = (const int*)d_in[4];
  const int*   col        = (const int*)d_in[5];
  const float* deg_emb    = (const float*)d_in[6];
  const float* deg_w      = (const float*)d_in[7];
  const float* deg_b      = (const float*)d_in[8];
  const float* merge_w    = (const float*)d_in[9];
  const float* merge_b    = (const float*)d_in[10];
  const float* tree_w     = (const float*)d_in[11];
  const float* tree_b     = (const float*)d_in[12];
  const float* pe_w       = (const float*)d_in[13];
  const float* pe_b       = (const float*)d_in[14];

  const int n_clique = in_sizes[0] / 64;
  const int n_edge   = in_sizes[4];

  float* agg16 = (float*)d_ws;                   // [n_clique, 16]
  float* cnt   = agg16 + (size_t)n_clique * 16;  // [n_clique]

  // 1) zero the aggregation workspace (17 floats per clique)
  const size_t nz = (size_t)n_clique * 17;
  zero_ws_kernel<<<(unsigned)((nz + 511) / 512), 512, 0, stream>>>(agg16, nz);

  // 2) edge gather/scatter-add of raw 16-dim LPE (16 lanes per edge)
  const size_t et = (size_t)n_edge * 16;
  edge_scatter_kernel<<<(unsigned)((et + 255) / 256), 256, 0, stream>>>(
      row, col, graph_lpe, agg16, cnt, n_edge);

  // 3) fused per-clique WMMA kernel (4 waves/block, 16 rows/wave)
  const int rows_per_block = WAVES_PER_BLOCK * TILE_M;
  const int blocks = (n_clique + rows_per_block - 1) / rows_per_block;
  clique_fused_kernel<<<blocks, WAVES_PER_BLOCK * 32, 0, stream>>>(
      x_clique, tree_deg, tree_lpe, deg_emb, deg_w, deg_b, merge_w, merge_b,
      tree_w, tree_b, pe_w, pe_b, agg16, cnt, (float*)d_out, n_clique);
}